// AGDN_40587440947768
// MI455X (gfx1250) — compile-verified
//
#include <hip/hip_runtime.h>
#include <hip/hip_bf16.h>

typedef __attribute__((ext_vector_type(2))) float v2f;
typedef __attribute__((ext_vector_type(8))) float v8f;

#define NEG_SLOPE 0.2f
#define EPSV 1e-16f

// ---- monotonic uint encoding of float for atomic max (handles negatives) ----
__device__ __forceinline__ unsigned fkey(float x) {
    unsigned u = __float_as_uint(x);
    return (u & 0x80000000u) ? ~u : (u | 0x80000000u);
}
__device__ __forceinline__ float funkey(unsigned k) {
    unsigned u = (k & 0x80000000u) ? (k & 0x7FFFFFFFu) : ~k;
    return __uint_as_float(u);
}
#define NEG_INF_KEY 0x007FFFFFu  // fkey(-inf)

// =====================  WMMA f32 GEMM: Y[M,NC] = X[M,K] @ W[K,NC]  ==========
// One wave per 16x16 output tile; blockDim.x = 32*(NC/16); gridDim.x = ceil(M/16).
__global__ void gemm16_wmma_f32(const float* __restrict__ X, const float* __restrict__ W,
                                float* __restrict__ Y, int M, int K, int NC) {
    const int lane = threadIdx.x & 31;
    const int wv   = threadIdx.x >> 5;      // column tile within block
    const int half = lane >> 4;             // 0: K{0,1}, 1: K{2,3}
    const int l15  = lane & 15;
    const int rowBase = blockIdx.x << 4;
    const int colBase = wv << 4;
    int row = rowBase + l15;
    if (row >= M) row = M - 1;              // clamp: keep EXEC all-1s for WMMA
    const int col = colBase + l15;

    v8f acc = {};
    for (int kt = 0; kt < K; kt += 4) {
        const int k0 = kt + (half << 1);
        v2f a, b;
        a.x = X[(long)row * K + k0];
        a.y = X[(long)row * K + k0 + 1];
        b.x = W[(long)k0 * NC + col];
        b.y = W[(long)(k0 + 1) * NC + col];
        acc = __builtin_amdgcn_wmma_f32_16x16x4_f32(false, a, false, b,
                                                    (short)0, acc, false, false);
    }
    // C/D layout: VGPR j -> M = j (lanes 0-15) or 8+j (lanes 16-31), N = lane&15
    const int rbase = rowBase + (half << 3);
#pragma unroll
    for (int j = 0; j < 8; ++j) {
        const int r = rbase + j;
        if (r < M) Y[(long)r * NC + col] = acc[j];
    }
}

// ===== per-node attention scalars al/ar + init amax/denom; one wave per node =====
__global__ void node_scalars(const float* __restrict__ cur,
                             const float* __restrict__ attl, const float* __restrict__ attr,
                             float* __restrict__ al, float* __restrict__ ar,
                             unsigned* __restrict__ amax_key, float* __restrict__ denom,
                             int Nn, int H, int C) {
    const int lane = threadIdx.x & 31;
    const int n = blockIdx.x * (blockDim.x >> 5) + (threadIdx.x >> 5);
    if (n >= Nn) return;
    const int F = H * C;
    for (int h = 0; h < H; ++h) {
        float sl = 0.f, sr = 0.f;
        for (int c = lane; c < C; c += 32) {
            const float v = cur[(long)n * F + h * C + c];
            sl += v * attl[h * C + c];
            sr += v * attr[h * C + c];
        }
#pragma unroll
        for (int off = 16; off > 0; off >>= 1) {
            sl += __shfl_xor(sl, off, 32);
            sr += __shfl_xor(sr, off, 32);
        }
        if (lane == 0) {
            al[n * H + h] = sl;
            ar[n * H + h] = sr;
            amax_key[n * H + h] = NEG_INF_KEY;
            denom[n * H + h] = 0.f;
        }
    }
}

// ===== pass 1: logits + segment max (atomic max on encoded key) =====
__global__ void edge_attn_max(const int* __restrict__ src, const int* __restrict__ dst,
                              const float* __restrict__ al, const float* __restrict__ ar,
                              float* __restrict__ logits, unsigned* __restrict__ amax_key,
                              int En, int H) {
    const int e = blockIdx.x * blockDim.x + threadIdx.x;
    if (e >= En) return;
    const int s = src[e], d = dst[e];
    for (int h = 0; h < H; ++h) {
        float a = al[s * H + h] + ar[d * H + h];
        a = a > 0.f ? a : NEG_SLOPE * a;
        logits[(long)e * H + h] = a;
        atomicMax(&amax_key[d * H + h], fkey(a));
    }
}

// decode encoded max keys back to float, in place
__global__ void decode_amax(float* buf, int n) {
    const int i = blockIdx.x * blockDim.x + threadIdx.x;
    if (i < n) buf[i] = funkey(__float_as_uint(buf[i]));
}

// ===== pass 2: exp(a - amax) and segment denom sum =====
__global__ void edge_exp_sum(const int* __restrict__ dst, const float* __restrict__ amax,
                             float* __restrict__ logits, float* __restrict__ denom,
                             int En, int H) {
    const int e = blockIdx.x * blockDim.x + threadIdx.x;
    if (e >= En) return;
    const int d = dst[e];
    for (int h = 0; h < H; ++h) {
        const float ex = __expf(logits[(long)e * H + h] - amax[d * H + h]);
        logits[(long)e * H + h] = ex;
        atomicAdd(&denom[d * H + h], ex);
    }
}

// ===== pass 3: weighted scatter-add; one wave per edge =====
__global__ void aggregate_edges(const int* __restrict__ src, const int* __restrict__ dst,
                                const float* __restrict__ logits, const float* __restrict__ denom,
                                const float* __restrict__ cur, float* __restrict__ outb,
                                int En, int H, int C) {
    const int lane = threadIdx.x & 31;
    const long e = (long)blockIdx.x * (blockDim.x >> 5) + (threadIdx.x >> 5);
    if (e >= En) return;
    const int s = src[e], d = dst[e];
    const float a0 = logits[e * H] / (denom[d * H] + EPSV);
    const float a1 = (H == 2) ? logits[e * H + 1] / (denom[d * H + 1] + EPSV) : a0;
    const int F = H * C;
    for (int f = lane; f < F; f += 32) {
        const float w = (f < C) ? a0 : a1;
        atomicAdd(&outb[(long)d * F + f], w * cur[(long)s * F + f]);
    }
}

// ===== elementwise helpers =====
__global__ void zero_f32(float* p, long n) {
    for (long i = blockIdx.x * (long)blockDim.x + threadIdx.x; i < n;
         i += (long)gridDim.x * blockDim.x) p[i] = 0.f;
}
__global__ void axpy_theta(float* __restrict__ acc, const float* __restrict__ rep,
                           const float* __restrict__ theta_row, long n, int C) {
    const int cm = C - 1;
    for (long i = blockIdx.x * (long)blockDim.x + threadIdx.x; i < n;
         i += (long)gridDim.x * blockDim.x)
        acc[i] += theta_row[(int)(i & cm)] * rep[i];
}
__global__ void finish_layer1(float* __restrict__ acc, const float* __restrict__ xres,
                              const float* __restrict__ bias, long n, int F) {
    const int fm = F - 1;
    for (long i = blockIdx.x * (long)blockDim.x + threadIdx.x; i < n;
         i += (long)gridDim.x * blockDim.x) {
        const float v = acc[i] + xres[i] + bias[(int)(i & fm)];
        acc[i] = v > 0.f ? v : (__expf(v) - 1.f);  // elu
    }
}
__global__ void init_out2(float* __restrict__ out, const float* __restrict__ res,
                          const float* __restrict__ bias, long n, int C) {
    const int cm = C - 1;
    for (long i = blockIdx.x * (long)blockDim.x + threadIdx.x; i < n;
         i += (long)gridDim.x * blockDim.x)
        out[i] = res[i] + bias[(int)(i & cm)];
}

// =====================  host orchestration  =====================
static void propagate_hop(const int* src, const int* dst,
                          const float* attl, const float* attr,
                          const float* cur, float* nxt,
                          float* AL, float* AR, float* AMAX, float* DEN, float* LOG,
                          int Nn, int En, int H, int C, hipStream_t stream) {
    const int F = H * C;
    node_scalars<<<(Nn + 7) / 8, 256, 0, stream>>>(cur, attl, attr, AL, AR,
                                                   (unsigned*)AMAX, DEN, Nn, H, C);
    edge_attn_max<<<(En + 255) / 256, 256, 0, stream>>>(src, dst, AL, AR, LOG,
                                                        (unsigned*)AMAX, En, H);
    decode_amax<<<(Nn * H + 255) / 256, 256, 0, stream>>>(AMAX, Nn * H);
    edge_exp_sum<<<(En + 255) / 256, 256, 0, stream>>>(dst, AMAX, LOG, DEN, En, H);
    zero_f32<<<4096, 256, 0, stream>>>(nxt, (long)Nn * F);
    aggregate_edges<<<(En + 7) / 8, 256, 0, stream>>>(src, dst, LOG, DEN, cur, nxt, En, H, C);
}

extern "C" void kernel_launch(void* const* d_in, const int* in_sizes, int n_in,
                              void* d_out, int out_size, void* d_ws, size_t ws_size,
                              hipStream_t stream) {
    const float* x     = (const float*)d_in[0];
    const int*   eidx  = (const int*)d_in[1];
    const float* W1    = (const float*)d_in[2];
    const float* attl1 = (const float*)d_in[3];
    const float* attr1 = (const float*)d_in[4];
    const float* th1   = (const float*)d_in[5];
    const float* b1    = (const float*)d_in[6];
    const float* W2    = (const float*)d_in[7];
    const float* attl2 = (const float*)d_in[8];
    const float* attr2 = (const float*)d_in[9];
    const float* th2   = (const float*)d_in[10];
    const float* b2    = (const float*)d_in[11];
    const float* rW2   = (const float*)d_in[12];

    const int Nn = in_sizes[0] / 128;       // 50000
    const int En = in_sizes[1] / 2;         // 1600000
    const int* src = eidx;
    const int* dst = eidx + En;

    // workspace layout (floats)
    float* ws  = (float*)d_ws;
    float* A   = ws;                         // [N,128] x_lin1 / rep0; later XL2|RES2
    float* B   = A   + (size_t)Nn * 128;     // [N,128] hop1
    float* C2  = B   + (size_t)Nn * 128;     // [N,128] hop2
    float* ACC = C2  + (size_t)Nn * 128;     // [N,128] layer1 accum -> h (after elu)
    float* AL  = ACC + (size_t)Nn * 128;     // [N,2]
    float* AR  = AL  + (size_t)Nn * 2;
    float* AMX = AR  + (size_t)Nn * 2;
    float* DEN = AMX + (size_t)Nn * 2;
    float* LOG = DEN + (size_t)Nn * 2;       // [E,2]

    float* out = (float*)d_out;              // [N,64]

    const int rowTiles = (Nn + 15) / 16;

    // ---------------- layer 1: heads=2, C=64 ----------------
    gemm16_wmma_f32<<<rowTiles, 256, 0, stream>>>(x, W1, A, Nn, 128, 128);
    propagate_hop(src, dst, attl1, attr1, A, B,  AL, AR, AMX, DEN, LOG, Nn, En, 2, 64, stream);
    propagate_hop(src, dst, attl1, attr1, B, C2, AL, AR, AMX, DEN, LOG, Nn, En, 2, 64, stream);
    zero_f32<<<4096, 256, 0, stream>>>(ACC, (long)Nn * 128);
    axpy_theta<<<4096, 256, 0, stream>>>(ACC, A,  th1 + 0 * 64, (long)Nn * 128, 64);
    axpy_theta<<<4096, 256, 0, stream>>>(ACC, B,  th1 + 1 * 64, (long)Nn * 128, 64);
    axpy_theta<<<4096, 256, 0, stream>>>(ACC, C2, th1 + 2 * 64, (long)Nn * 128, 64);
    finish_layer1<<<4096, 256, 0, stream>>>(ACC, A, b1, (long)Nn * 128, 128); // ACC = elu(h)

    // ---------------- layer 2: heads=1, C=64 ----------------
    float* XL2  = A;                         // [N,64]
    float* RES2 = A + (size_t)Nn * 64;       // [N,64]
    gemm16_wmma_f32<<<rowTiles, 128, 0, stream>>>(ACC, W2,  XL2,  Nn, 128, 64);
    gemm16_wmma_f32<<<rowTiles, 128, 0, stream>>>(ACC, rW2, RES2, Nn, 128, 64);
    init_out2<<<4096, 256, 0, stream>>>(out, RES2, b2, (long)Nn * 64, 64);
    axpy_theta<<<4096, 256, 0, stream>>>(out, XL2, th2 + 0 * 64, (long)Nn * 64, 64);
    propagate_hop(src, dst, attl2, attr2, XL2, B, AL, AR, AMX, DEN, LOG, Nn, En, 1, 64, stream);
    axpy_theta<<<4096, 256, 0, stream>>>(out, B, th2 + 1 * 64, (long)Nn * 64, 64);
    propagate_hop(src, dst, attl2, attr2, B, C2, AL, AR, AMX, DEN, LOG, Nn, En, 1, 64, stream);
    axpy_theta<<<4096, 256, 0, stream>>>(out, C2, th2 + 2 * 64, (long)Nn * 64, 64);
}